// ModelP_46394236732088
// MI455X (gfx1250) — compile-verified
//
#include <hip/hip_runtime.h>
#include <cstdint>
#include <cstddef>

#define DD 128
#define NEG_SLOPE 0.01f

typedef __attribute__((ext_vector_type(2))) float v2f;
typedef __attribute__((ext_vector_type(8))) float v8f;

// ---------------------------------------------------------------------------
// GEMM: C[M x 128] = A[M x 128] @ W[128 x 128] (+bias) (+activation)
// mode: 0 = none, 1 = bias+leaky_relu, 2 = bias+relu
// One block = 8 waves; block owns a 16-row strip of A (staged in LDS),
// wave w owns output columns [16w, 16w+16). K-loop: 32 x v_wmma_f32_16x16x4_f32.
// ---------------------------------------------------------------------------
__global__ __launch_bounds__(256) void gemm128_wmma(
    const float* __restrict__ A, const float* __restrict__ W,
    const float* __restrict__ bias, float* __restrict__ C, int M, int mode)
{
    __shared__ float As[16][132];   // pad 128->132 to spread LDS banks
    const int tid  = threadIdx.x;
    const int row0 = blockIdx.x << 4;

    for (int idx = tid; idx < 16 * DD; idx += 256) {
        int r = idx >> 7, c = idx & 127;
        int gr = row0 + r;
        As[r][c] = (gr < M) ? A[(size_t)gr * DD + c] : 0.0f;
    }
    __syncthreads();

    const int lane = tid & 31;
    const int wave = tid >> 5;
    const int col0 = wave << 4;
    const int m    = lane & 15;
    const int koff = (lane >> 4) << 1;   // lanes 0-15: K {k,k+1}; lanes 16-31: {k+2,k+3}

    v8f acc = {};
    for (int k0 = 0; k0 < DD; k0 += 4) {
        v2f a, b;
        a.x = As[m][k0 + koff];
        a.y = As[m][k0 + koff + 1];
        b.x = W[(size_t)(k0 + koff)     * DD + col0 + m];
        b.y = W[(size_t)(k0 + koff + 1) * DD + col0 + m];
        acc = __builtin_amdgcn_wmma_f32_16x16x4_f32(
            false, a, false, b, (short)0, acc, false, false);
    }

    const int   colg  = col0 + m;
    const float bv    = bias ? bias[colg] : 0.0f;
    const int   rbase = row0 + ((lane < 16) ? 0 : 8);
#pragma unroll
    for (int r = 0; r < 8; ++r) {
        int gr = rbase + r;
        if (gr < M) {
            float v = acc[r] + bv;
            if (mode == 1)      v = (v > 0.0f) ? v : NEG_SLOPE * v;
            else if (mode == 2) v = fmaxf(v, 0.0f);
            C[(size_t)gr * DD + colg] = v;
        }
    }
}

// ---------------------------------------------------------------------------
// deg[dst] += 1 for each valid edge (src,dst >= 0)
// ---------------------------------------------------------------------------
__global__ void deg_kernel(const int* __restrict__ src, const int* __restrict__ dst,
                           float* __restrict__ deg, int E)
{
    int e = blockIdx.x * blockDim.x + threadIdx.x;
    if (e >= E) return;
    int s = src[e], d = dst[e];
    if (s < 0 || d < 0) return;
    unsafeAtomicAdd(&deg[d], 1.0f);
}

// dinv[i] = rsqrt(deg[i] + 1)   (in place)
__global__ void rsqrt_kernel(float* __restrict__ deg, int M)
{
    int i = blockIdx.x * blockDim.x + threadIdx.x;
    if (i >= M) return;
    deg[i] = rsqrtf(deg[i] + 1.0f);
}

// ---------------------------------------------------------------------------
// Edge scatter: agg[dst] += xw[src] * norm, norm = dinv[s]*dinv[d] (or 1).
// One wave32 per edge, float4 per lane (128 floats per edge).
// ---------------------------------------------------------------------------
__global__ __launch_bounds__(256) void edge_agg(
    const float4* __restrict__ xw, float* __restrict__ agg,
    const int* __restrict__ src, const int* __restrict__ dst,
    const float* __restrict__ dinv, int E, int useNorm)
{
    int w    = (blockIdx.x * blockDim.x + threadIdx.x) >> 5;
    int lane = threadIdx.x & 31;
    if (w >= E) return;
    int s = src[w], d = dst[w];
    if (s < 0 || d < 0) return;
    float norm = useNorm ? dinv[s] * dinv[d] : 1.0f;
    float4 v = xw[(size_t)s * 32 + lane];
    float* ap = agg + (size_t)d * DD + lane * 4;
    unsafeAtomicAdd(ap + 0, v.x * norm);
    unsafeAtomicAdd(ap + 1, v.y * norm);
    unsafeAtomicAdd(ap + 2, v.z * norm);
    unsafeAtomicAdd(ap + 3, v.w * norm);
}

// h_out = leaky(agg + xw * dinv[row]^2 + bias)
__global__ void combine_kernel(const float4* __restrict__ agg,
                               const float4* __restrict__ xw,
                               const float* __restrict__ bias,
                               const float* __restrict__ dinv,
                               float4* __restrict__ out, int M)
{
    int t = blockIdx.x * blockDim.x + threadIdx.x;
    int node = t >> 5, q = t & 31;
    if (node >= M) return;
    float di = dinv[node];
    float self = di * di;
    float4 a  = agg[(size_t)node * 32 + q];
    float4 w  = xw [(size_t)node * 32 + q];
    float4 bv = ((const float4*)bias)[q];
    float4 v;
    v.x = a.x + w.x * self + bv.x;
    v.y = a.y + w.y * self + bv.y;
    v.z = a.z + w.z * self + bv.z;
    v.w = a.w + w.w * self + bv.w;
    v.x = (v.x > 0.0f) ? v.x : NEG_SLOPE * v.x;
    v.y = (v.y > 0.0f) ? v.y : NEG_SLOPE * v.y;
    v.z = (v.z > 0.0f) ? v.z : NEG_SLOPE * v.z;
    v.w = (v.w > 0.0f) ? v.w : NEG_SLOPE * v.w;
    out[(size_t)node * 32 + q] = v;
}

// score[i] = <h[i], Wp_root> + <aggp[i], Wp_rel> + b_pool   (one wave per node)
__global__ void score_kernel(const float4* __restrict__ h,
                             const float4* __restrict__ aggp,
                             const float* __restrict__ wroot,
                             const float* __restrict__ wrel,
                             const float* __restrict__ bpool,
                             float* __restrict__ score, int M)
{
    int w    = (blockIdx.x * blockDim.x + threadIdx.x) >> 5;
    int lane = threadIdx.x & 31;
    if (w >= M) return;
    float4 hv = h   [(size_t)w * 32 + lane];
    float4 av = aggp[(size_t)w * 32 + lane];
    float4 wr = ((const float4*)wroot)[lane];
    float4 we = ((const float4*)wrel )[lane];
    float p = hv.x * wr.x + hv.y * wr.y + hv.z * wr.z + hv.w * wr.w
            + av.x * we.x + av.y * we.y + av.z * we.z + av.w * we.w;
#pragma unroll
    for (int off = 16; off > 0; off >>= 1) p += __shfl_down(p, off, 32);
    if (lane == 0) score[w] = p + bpool[0];
}

// Exact descending rank with stable index tie-break (matches jax.lax.top_k).
__global__ __launch_bounds__(256) void rank_kernel(const float* __restrict__ score,
                                                   int* __restrict__ rank, int N)
{
    __shared__ float sc[256];
    int i = blockIdx.x * 256 + threadIdx.x;
    float si = (i < N) ? score[i] : 0.0f;
    int cnt = 0;
    for (int base = 0; base < N; base += 256) {
        int j = base + threadIdx.x;
        sc[threadIdx.x] = (j < N) ? score[j] : -__builtin_inff();
        __syncthreads();
        int lim = min(256, N - base);
        for (int t = 0; t < lim; ++t) {
            float sj = sc[t];
            int   j2 = base + t;
            cnt += ((sj > si) || (sj == si && j2 < i)) ? 1 : 0;
        }
        __syncthreads();
    }
    if (i < N) rank[i] = cnt;
}

// hp[rank[i]] = h[i] * tanh(score[i])  for selected nodes (one wave per node)
__global__ void gather_kernel(const float4* __restrict__ h,
                              const int* __restrict__ rank,
                              const float* __restrict__ score,
                              float4* __restrict__ hp, int N, int K)
{
    int w    = (blockIdx.x * blockDim.x + threadIdx.x) >> 5;
    int lane = threadIdx.x & 31;
    if (w >= N) return;
    int r = rank[w];
    if (r >= K) return;
    float t = tanhf(score[w]);
    float4 v = h[(size_t)w * 32 + lane];
    v.x *= t; v.y *= t; v.z *= t; v.w *= t;
    hp[(size_t)r * 32 + lane] = v;
}

// Relabel edges to pooled ids; dropped edges -> (-1,-1)
__global__ void remap_kernel(const int* __restrict__ src, const int* __restrict__ dst,
                             const int* __restrict__ rank,
                             int* __restrict__ nsrc, int* __restrict__ ndst, int E, int K)
{
    int e = blockIdx.x * blockDim.x + threadIdx.x;
    if (e >= E) return;
    int rs = rank[src[e]], rd = rank[dst[e]];
    bool ok = (rs < K) && (rd < K);
    nsrc[e] = ok ? rs : -1;
    ndst[e] = ok ? rd : -1;
}

// ---------------------------------------------------------------------------
// Host side
// ---------------------------------------------------------------------------
static void run_gcn(const float* h, const float* Wm, const float* b,
                    const int* s, const int* d, const float* dinv,
                    float* xw, float* agg, float* hout,
                    int M, int E, hipStream_t st)
{
    gemm128_wmma<<<(M + 15) / 16, 256, 0, st>>>(h, Wm, nullptr, xw, M, 0);
    (void)hipMemsetAsync(agg, 0, (size_t)M * DD * sizeof(float), st);
    edge_agg<<<(E * 32 + 255) / 256, 256, 0, st>>>((const float4*)xw, agg, s, d, dinv, E, 1);
    combine_kernel<<<(M * 32 + 255) / 256, 256, 0, st>>>(
        (const float4*)agg, (const float4*)xw, b, dinv, (float4*)hout, M);
}

extern "C" void kernel_launch(void* const* d_in, const int* in_sizes, int n_in,
                              void* d_out, int out_size, void* d_ws, size_t ws_size,
                              hipStream_t stream)
{
    const float* x       = (const float*)d_in[0];
    const int*   ei      = (const int*)  d_in[1];
    const float* W_lin1  = (const float*)d_in[2];
    const float* b_lin1  = (const float*)d_in[3];
    const float* W1      = (const float*)d_in[4];
    const float* b1      = (const float*)d_in[5];
    const float* W2      = (const float*)d_in[6];
    const float* b2      = (const float*)d_in[7];
    const float* Wp_root = (const float*)d_in[8];
    const float* Wp_rel  = (const float*)d_in[9];
    const float* b_pool  = (const float*)d_in[10];
    const float* W3      = (const float*)d_in[11];
    const float* b3      = (const float*)d_in[12];
    const float* W4      = (const float*)d_in[13];
    const float* b4      = (const float*)d_in[14];
    const float* W5      = (const float*)d_in[15];
    const float* b5      = (const float*)d_in[16];
    const float* W6      = (const float*)d_in[17];
    const float* b6      = (const float*)d_in[18];
    const float* W_lin2  = (const float*)d_in[19];
    const float* b_lin2  = (const float*)d_in[20];

    const int N = in_sizes[0] / DD;     // 50000
    const int E = in_sizes[1] / 2;      // 800000
    const int K = out_size / DD;        // 25000
    const int* src = ei;
    const int* dst = ei + E;

    // Workspace carve-out (256B aligned)
    size_t off = 0;
    auto take = [&](size_t bytes) -> void* {
        off = (off + 255) & ~(size_t)255;
        void* r = (char*)d_ws + off;
        off += bytes;
        return r;
    };
    float* hA    = (float*)take((size_t)N * DD * 4);
    float* xw    = (float*)take((size_t)N * DD * 4);
    float* agg   = (float*)take((size_t)N * DD * 4);
    float* hB    = (float*)take((size_t)K * DD * 4);
    float* dinvN = (float*)take((size_t)N * 4);
    float* dinvK = (float*)take((size_t)K * 4);
    float* score = (float*)take((size_t)N * 4);
    int*   rank  = (int*)  take((size_t)N * 4);
    int*   nsrc  = (int*)  take((size_t)E * 4);
    int*   ndst  = (int*)  take((size_t)E * 4);

    const int eT  = (E + 255) / 256;           // thread-per-edge grids
    const int nT  = (N + 255) / 256;
    const int kT  = (K + 255) / 256;
    const int eW  = (E * 32 + 255) / 256;      // wave-per-edge grids
    const int nW  = (N * 32 + 255) / 256;

    // h = leaky(x @ W_lin1 + b_lin1)
    gemm128_wmma<<<(N + 15) / 16, 256, 0, stream>>>(x, W_lin1, b_lin1, hA, N, 1);

    // degrees / dinv for the full graph
    (void)hipMemsetAsync(dinvN, 0, (size_t)N * 4, stream);
    deg_kernel<<<eT, 256, 0, stream>>>(src, dst, dinvN, E);
    rsqrt_kernel<<<nT, 256, 0, stream>>>(dinvN, N);

    // two GCN layers on the full graph
    run_gcn(hA, W1, b1, src, dst, dinvN, xw, agg, hA, N, E, stream);
    run_gcn(hA, W2, b2, src, dst, dinvN, xw, agg, hA, N, E, stream);

    // SAGPool scoring: unnormalized neighbor sum + two dot products
    (void)hipMemsetAsync(agg, 0, (size_t)N * DD * 4, stream);
    edge_agg<<<eW, 256, 0, stream>>>((const float4*)hA, agg, src, dst, dinvN, E, 0);
    score_kernel<<<nW, 256, 0, stream>>>((const float4*)hA, (const float4*)agg,
                                         Wp_root, Wp_rel, b_pool, score, N);

    // exact top-K rank, gather+scale, edge relabel
    rank_kernel<<<nT, 256, 0, stream>>>(score, rank, N);
    gather_kernel<<<nW, 256, 0, stream>>>((const float4*)hA, rank, score, (float4*)hB, N, K);
    remap_kernel<<<eT, 256, 0, stream>>>(src, dst, rank, nsrc, ndst, E, K);

    // degrees / dinv for the pooled graph
    (void)hipMemsetAsync(dinvK, 0, (size_t)K * 4, stream);
    deg_kernel<<<eT, 256, 0, stream>>>(nsrc, ndst, dinvK, E);
    rsqrt_kernel<<<kT, 256, 0, stream>>>(dinvK, K);

    // four GCN layers on the pooled graph
    run_gcn(hB, W3, b3, nsrc, ndst, dinvK, xw, agg, hB, K, E, stream);
    run_gcn(hB, W4, b4, nsrc, ndst, dinvK, xw, agg, hB, K, E, stream);
    run_gcn(hB, W5, b5, nsrc, ndst, dinvK, xw, agg, hB, K, E, stream);
    run_gcn(hB, W6, b6, nsrc, ndst, dinvK, xw, agg, hB, K, E, stream);

    // out = relu(hB @ W_lin2 + b_lin2)
    gemm128_wmma<<<(K + 15) / 16, 256, 0, stream>>>(hB, W_lin2, b_lin2, (float*)d_out, K, 2);
}